// Loss_81501299409301
// MI455X (gfx1250) — compile-verified
//
#include <hip/hip_runtime.h>
#include <stdint.h>

#define THRESH   0.35f
#define FTHRESH  0.2f
#define VPT      8            // priors per thread in the best-prior kernel

typedef __attribute__((ext_vector_type(2))) float v2f;
typedef __attribute__((ext_vector_type(8))) float v8f;
typedef unsigned int u32x4 __attribute__((ext_vector_type(4)));
typedef int          i32x4 __attribute__((ext_vector_type(4)));
typedef int          i32x8 __attribute__((ext_vector_type(8)));

// ---------- helpers ----------

__device__ __forceinline__ unsigned long long pack_ov(float iou, int p) {
  // iou >= 0 so float bits are monotonic; invert index so ties pick smallest p
  return ((unsigned long long)__float_as_uint(iou) << 32) |
         (unsigned long long)(0x7FFFFFFFu - (unsigned)p);
}

__device__ __forceinline__ unsigned long long shflxor_u64(unsigned long long v, int m) {
  unsigned lo = (unsigned)v, hi = (unsigned)(v >> 32);
  lo = __shfl_xor(lo, m, 32);
  hi = __shfl_xor(hi, m, 32);
  return ((unsigned long long)hi << 32) | lo;
}

__device__ __forceinline__ float iou_box(float ax1, float ay1, float ax2, float ay2,
                                         float bx1, float by1, float bx2, float by2,
                                         float areaB) {
  float ltx = fmaxf(ax1, bx1), lty = fmaxf(ay1, by1);
  float rbx = fminf(ax2, bx2), rby = fminf(ay2, by2);
  float w = fmaxf(rbx - ltx, 0.0f), h = fmaxf(rby - lty, 0.0f);
  float inter = w * h;
  float areaA = (ax2 - ax1) * (ay2 - ay1);
  return inter / (areaA + areaB - inter);
}

__device__ __forceinline__ float smooth_l1(float x) {
  float ax = fabsf(x);
  return ax < 1.0f ? 0.5f * x * x : ax - 0.5f;
}

// TDM: DMA targets[b, 0:64, 0:5] (label + box) into LDS as a 2D tile.
// Tile 5x64 dwords, row stride 15 dwords -> compact stride-5 rows in LDS.
// Tracked on TENSORcnt. Issue from one wave only.
__device__ __forceinline__ void tdm_load_truths(const float* tb, float* sTB) {
  unsigned long long ga = (unsigned long long)(uintptr_t)tb;
  u32x4 g0;
  g0[0] = 1u;                                   // count=1 valid descriptor
  g0[1] = (unsigned)(uintptr_t)(void*)sTB;      // lds_addr (low 32 bits of generic ptr)
  g0[2] = (unsigned)ga;                         // global_addr[31:0]
  g0[3] = (unsigned)((ga >> 32) & 0x01FFFFFFull) | 0x80000000u; // addr[56:32] | type=2
  i32x8 g1;
  g1[0] = 0x00020000;   // data_size=2 (4 bytes)
  g1[1] = 15 << 16;     // tensor_dim0 = 15 (row length for OOB)
  g1[2] = 64 << 16;     // tensor_dim1 = 64 rows
  g1[3] = 5 << 16;      // tile_dim0 = 5 dwords per row
  g1[4] = 64;           // tile_dim1 = 64 rows, tile_dim2 = 0
  g1[5] = 15;           // tensor_dim0_stride = 15 dwords
  g1[6] = 0;
  g1[7] = 0;
  i32x4 gz = {0, 0, 0, 0};
#if defined(__clang_major__) && (__clang_major__ >= 23)
  i32x8 gz8 = {0, 0, 0, 0, 0, 0, 0, 0};
  __builtin_amdgcn_tensor_load_to_lds(g0, g1, gz, gz, gz8, 0);
#else
  __builtin_amdgcn_tensor_load_to_lds(g0, g1, gz, gz, 0);
#endif
}

// 32-lane sum via V_WMMA_F32_16X16X4_F32 with B = ones:
// A[m][0]=v_m (lanes 0-15), A[m][2]=v_{m+16} (lanes 16-31), K1/K3 = 0.
// D[m][n] = v_m + v_{m+16}; sum 8 D VGPRs -> half-sums, combine with shfl_xor(16).
__device__ __forceinline__ float wave_sum32(float v) {
  v2f a; a.x = v;    a.y = 0.0f;
  v2f o; o.x = 1.0f; o.y = 1.0f;
  v8f c = {};
  c = __builtin_amdgcn_wmma_f32_16x16x4_f32(false, a, false, o, (short)0, c, false, false);
  float t = c[0] + c[1] + c[2] + c[3] + c[4] + c[5] + c[6] + c[7];
  t += __shfl_xor(t, 16, 32);
  return t;
}

// ---------- kernel 0: init workspace ----------

__global__ void init_kernel(unsigned long long* bp_packed, float* acc, int n) {
  int i = blockIdx.x * 256 + threadIdx.x;
  if (i < n) bp_packed[i] = 0ull;
  if (i < 3) acc[i] = 0.0f;
}

// ---------- kernel A: per-truth max over priors (packed overlap|idx) ----------
// 8 priors/thread: local register max before the wave butterfly cuts
// ds_bpermute + atomicMax traffic 8x vs one prior/thread.

__global__ void __launch_bounds__(256)
best_prior_kernel(const float* __restrict__ priors,
                  const float* __restrict__ targets,
                  unsigned long long* __restrict__ bp_packed, int P) {
  __shared__ float sTB[64 * 5];
  __shared__ unsigned long long sWM[8 * 64];
  const int b = blockIdx.y;
  const int tid = threadIdx.x;
  const int lane = tid & 31, wave = tid >> 5;

  if (tid < 32) {
    tdm_load_truths(targets + (size_t)b * 64 * 15, sTB);
    __builtin_amdgcn_s_wait_tensorcnt(0);
  }
  __syncthreads();

  const int p0 = blockIdx.x * (256 * VPT) + tid;
  float bx1[VPT], by1[VPT], bx2[VPT], by2[VPT], areaB[VPT];
  int pidx[VPT];
  #pragma unroll
  for (int j = 0; j < VPT; ++j) {
    int p = p0 + j * 256;
    bool act = p < P;
    pidx[j] = act ? p : -1;
    float4 pr = ((const float4*)priors)[act ? p : 0];
    bx1[j] = pr.x - pr.z * 0.5f;  by1[j] = pr.y - pr.w * 0.5f;
    bx2[j] = pr.x + pr.z * 0.5f;  by2[j] = pr.y + pr.w * 0.5f;
    areaB[j] = pr.z * pr.w;
  }

  for (int o = 0; o < 64; ++o) {
    float ax1 = sTB[o * 5 + 1], ay1 = sTB[o * 5 + 2];
    float ax2 = sTB[o * 5 + 3], ay2 = sTB[o * 5 + 4];
    unsigned long long pk = 0ull;
    #pragma unroll
    for (int j = 0; j < VPT; ++j) {
      float iou = iou_box(ax1, ay1, ax2, ay2, bx1[j], by1[j], bx2[j], by2[j], areaB[j]);
      unsigned long long c = (pidx[j] >= 0) ? pack_ov(iou, pidx[j]) : 0ull;
      if (c > pk) pk = c;
    }
    #pragma unroll
    for (int m = 16; m >= 1; m >>= 1) {
      unsigned long long o2 = shflxor_u64(pk, m);
      pk = (o2 > pk) ? o2 : pk;
    }
    if (lane == 0) sWM[wave * 64 + o] = pk;
  }
  __syncthreads();
  if (tid < 64) {
    unsigned long long best = 0ull;
    #pragma unroll
    for (int w = 0; w < 8; ++w) {
      unsigned long long v = sWM[w * 64 + tid];
      if (v > best) best = v;
    }
    atomicMax(&bp_packed[(size_t)b * 64 + tid], best);
  }
}

// ---------- kernel B: per-prior match, encode, losses ----------

__global__ void __launch_bounds__(256)
loss_kernel(const float* __restrict__ loc_data,
            const float* __restrict__ conf_data,
            const float* __restrict__ priors,
            const float* __restrict__ targets,
            const unsigned long long* __restrict__ bp_packed,
            float* __restrict__ acc, int P) {
  __shared__ float    sTB[64 * 5];
  __shared__ unsigned sPK[128];
  __shared__ int      sBPI[64];
  __shared__ int      sVAL[64];
  __shared__ int      sAnyV;
  __shared__ float    sWr[8][3];
  const int b = blockIdx.y;
  const int tid = threadIdx.x;
  const int lane = tid & 31, wave = tid >> 5;

  if (tid < 32) {
    tdm_load_truths(targets + (size_t)b * 64 * 15, sTB);
    // packed best-prior table via async global->LDS (ASYNCcnt path)
    uint32_t base = (uint32_t)(uintptr_t)(void*)sPK;
    const unsigned* g = (const unsigned*)(bp_packed + (size_t)b * 64);
    #pragma unroll
    for (int i = 0; i < 4; ++i) {
      int e = i * 32 + lane;   // 128 dwords = 64 packed u64, little-endian dword order
      asm volatile("global_load_async_to_lds_b32 %0, %1, off"
                   :: "v"(base + (uint32_t)(e * 4)), "v"(g + e) : "memory");
    }
    asm volatile("s_wait_asynccnt 0" ::: "memory");
    __builtin_amdgcn_s_wait_tensorcnt(0);
  }
  __syncthreads();

  if (tid < 64) {
    unsigned lo = sPK[tid * 2 + 0], hi = sPK[tid * 2 + 1];
    sBPI[tid] = (int)(0x7FFFFFFFu - lo);                    // best prior index for truth tid
    sVAL[tid] = (__uint_as_float(hi) >= FTHRESH) ? 1 : 0;   // valid_gt
  }
  __syncthreads();
  if (tid == 0) {
    int a = 0;
    for (int o = 0; o < 64; ++o) a |= sVAL[o];
    sAnyV = a;
  }
  __syncthreads();

  const int p = blockIdx.x * 256 + tid;
  const bool act = p < P;
  const float4 pr = ((const float4*)priors)[act ? p : 0];
  const float bx1 = pr.x - pr.z * 0.5f, by1 = pr.y - pr.w * 0.5f;
  const float bx2 = pr.x + pr.z * 0.5f, by2 = pr.y + pr.w * 0.5f;
  const float areaB = pr.z * pr.w;

  // best truth per prior (first-index argmax, matching jnp.argmax)
  float bto = -1.0f; int bti = 0;
  for (int o = 0; o < 64; ++o) {
    float ax1 = sTB[o * 5 + 1], ay1 = sTB[o * 5 + 2];
    float ax2 = sTB[o * 5 + 3], ay2 = sTB[o * 5 + 4];
    float iou = iou_box(ax1, ay1, ax2, ay2, bx1, by1, bx2, by2, areaB);
    if (iou > bto) { bto = iou; bti = o; }
  }
  // force-match fixup: fill uses PRE-update overlap at this prior (bto_orig); last o wins
  const float bto_orig = bto;
  for (int o = 0; o < 64; ++o) {
    if (sBPI[o] == p) { bto = sVAL[o] ? 2.0f : bto_orig; bti = o; }
  }

  int conf = 0;
  if (act && sAnyV) {
    conf = (int)sTB[bti * 5 + 0];          // label
    if (bto < THRESH) conf = 0;
  }

  float sl = 0.0f, sc = 0.0f, cnt = 0.0f;
  if (conf != 0) {
    float ax1 = sTB[bti * 5 + 1], ay1 = sTB[bti * 5 + 2];
    float ax2 = sTB[bti * 5 + 3], ay2 = sTB[bti * 5 + 4];
    float gcx = ((ax1 + ax2) * 0.5f - pr.x) / (0.1f * pr.z);
    float gcy = ((ay1 + ay2) * 0.5f - pr.y) / (0.1f * pr.w);
    float gw  = __logf((ax2 - ax1) / pr.z) * 5.0f;   // / VAR1 (0.2)
    float gh  = __logf((ay2 - ay1) / pr.w) * 5.0f;
    float4 ld = ((const float4*)loc_data)[(size_t)b * P + p];
    sl = smooth_l1(ld.x - gcx) + smooth_l1(ld.y - gcy) +
         smooth_l1(ld.z - gw)  + smooth_l1(ld.w - gh);
    float2 cd = ((const float2*)conf_data)[(size_t)b * P + p];
    float mx = fmaxf(cd.x, cd.y);
    float lse = mx + __logf(__expf(cd.x - mx) + __expf(cd.y - mx));
    int tgt = conf - 1; tgt = tgt < 0 ? 0 : (tgt > 1 ? 1 : tgt);
    sc = lse - (tgt ? cd.y : cd.x);
    cnt = 1.0f;
  }

  // EXEC is all-ones here (no early returns): WMMA-based 32-lane sums
  float wsl = wave_sum32(sl);
  float wsc = wave_sum32(sc);
  float wcn = wave_sum32(cnt);
  if (lane == 0) { sWr[wave][0] = wsl; sWr[wave][1] = wsc; sWr[wave][2] = wcn; }
  __syncthreads();
  if (tid == 0) {
    float a0 = 0, a1 = 0, a2 = 0;
    #pragma unroll
    for (int w = 0; w < 8; ++w) { a0 += sWr[w][0]; a1 += sWr[w][1]; a2 += sWr[w][2]; }
    atomicAdd(&acc[0], a0);
    atomicAdd(&acc[1], a1);
    atomicAdd(&acc[2], a2);
  }
}

// ---------- kernel C: finalize ----------

__global__ void final_kernel(const float* __restrict__ acc, float* __restrict__ out) {
  float n = fmaxf(acc[2], 1.0f);
  out[0] = acc[0] / n;
  out[1] = acc[1] / n;
}

// ---------- launch ----------

extern "C" void kernel_launch(void* const* d_in, const int* in_sizes, int n_in,
                              void* d_out, int out_size, void* d_ws, size_t ws_size,
                              hipStream_t stream) {
  const float* loc_data  = (const float*)d_in[0];
  const float* conf_data = (const float*)d_in[1];
  // d_in[2] = landm_data (unused by the reference loss)
  const float* priors    = (const float*)d_in[3];
  const float* targets   = (const float*)d_in[4];

  const int P = in_sizes[3] / 4;          // 28896
  const int B = in_sizes[4] / (64 * 15);  // 64

  float* acc = (float*)d_ws;                                        // 3 floats
  unsigned long long* bp_packed =
      (unsigned long long*)((char*)d_ws + 16);                      // B*64 u64

  const int nPK = B * 64;
  init_kernel<<<(nPK + 255) / 256, 256, 0, stream>>>(bp_packed, acc, nPK);

  const int tilesA = (P + 256 * VPT - 1) / (256 * VPT);
  dim3 gridA(tilesA, B);
  best_prior_kernel<<<gridA, 256, 0, stream>>>(priors, targets, bp_packed, P);

  const int tilesB = (P + 255) / 256;
  dim3 gridB(tilesB, B);
  loss_kernel<<<gridB, 256, 0, stream>>>(loc_data, conf_data, priors, targets,
                                         bp_packed, acc, P);
  final_kernel<<<1, 1, 0, stream>>>(acc, (float*)d_out);
}